// TopKAbsolutes2D_74079595922024
// MI455X (gfx1250) — compile-verified
//
#include <hip/hip_runtime.h>
#include <hip/hip_bf16.h>
#include <stdint.h>

// ---------------- problem constants ----------------
#define BATCH      32
#define N_PER_ROW  (64 * 224 * 224)          // 3,211,264
#define NV4        (N_PER_ROW / 4)           // 802,816 = 256 * 3136
#define THREADS    256

#define BINS1      32768                     // key bits [30:16]
#define BINS2      65536                     // key bits [15:0]

#define HBLOCKS_X  32
#define HITERS     (NV4 / (THREADS * HBLOCKS_X))   // 98
#define WBLOCKS_X  49
#define WITERS     (NV4 / (THREADS * WBLOCKS_X))   // 64
#define WUNROLL    4

#define PIPE       8                         // async-to-LDS ring depth (power of 2)

#define CAND_CAP   4096

// ---------------- workspace layout (bytes) ----------------
#define HIST1_OFF   0
#define HIST1_BYTES ((size_t)BATCH * BINS1 * 4)               // 4 MiB
#define HIST2_OFF   (HIST1_OFF + HIST1_BYTES)
#define HIST2_BYTES ((size_t)BATCH * BINS2 * 4)               // 8 MiB
#define META_OFF    (HIST2_OFF + HIST2_BYTES)                 // 12 MiB
#define CAND_OFF    (META_OFF + 4096)
#define CAND_BYTES  ((size_t)BATCH * CAND_CAP * 4)
#define WS_ZERO_BYTES (CAND_OFF + CAND_BYTES)

// per-row meta slots (8 u32 per row)
#define M_B1       0
#define M_K2       1
#define M_THR      2
#define M_KEEPEQ   3
#define M_CANDCNT  4

typedef float __attribute__((ext_vector_type(4))) f4;

// ---------------- CDNA5 async global->LDS helpers ----------------
// Per-lane: copies 16 bytes from the global address gaddr to the LDS address in lds_byte_addr. Tracked by ASYNCcnt.
__device__ __forceinline__ void async_ld_b128(uint32_t lds_byte_addr, const void* gaddr) {
    asm volatile("global_load_async_to_lds_b128 %0, %1, off"
                 :: "v"(lds_byte_addr), "v"(gaddr)
                 : "memory");
}
// Wait until at most n async ops outstanding (ASYNCcnt <= n). n must be small const-ish.
__device__ __forceinline__ void wait_async_le(int n) {
    switch (n) {
    case 0: asm volatile("s_wait_asynccnt 0x0" ::: "memory"); break;
    case 1: asm volatile("s_wait_asynccnt 0x1" ::: "memory"); break;
    case 2: asm volatile("s_wait_asynccnt 0x2" ::: "memory"); break;
    case 3: asm volatile("s_wait_asynccnt 0x3" ::: "memory"); break;
    case 4: asm volatile("s_wait_asynccnt 0x4" ::: "memory"); break;
    case 5: asm volatile("s_wait_asynccnt 0x5" ::: "memory"); break;
    case 6: asm volatile("s_wait_asynccnt 0x6" ::: "memory"); break;
    default: asm volatile("s_wait_asynccnt 0x7" ::: "memory"); break;
    }
}
// Flat shared pointer -> 32-bit LDS byte offset (flat LDS aperture keeps the
// wave-relative LDS offset in addr[31:0] per CDNA5 ISA flat addressing).
__device__ __forceinline__ uint32_t lds_addr_of(const void* p) {
    return (uint32_t)(uintptr_t)p;
}

__device__ __forceinline__ uint32_t abs_key(float f) {
    return __float_as_uint(f) & 0x7fffffffu;
}

// ---------------- kernel 1: coarse histogram (bits [30:16]) ----------------
__global__ void __launch_bounds__(THREADS)
hist1_kernel(const float* __restrict__ x, uint32_t* __restrict__ g_hist1) {
    __shared__ uint32_t hist[BINS1];          // 128 KB
    __shared__ f4 stage[PIPE][THREADS];       // 32 KB ring, wave-private 16B slots

    const int tid = threadIdx.x;
    const int row = blockIdx.y;

    for (int b = tid; b < BINS1; b += THREADS) hist[b] = 0;
    __syncthreads();

    const f4* __restrict__ src = (const f4*)x
        + (size_t)row * NV4 + (size_t)blockIdx.x * (THREADS * HITERS);

    const uint32_t lds_base = lds_addr_of(&stage[0][tid]);
    const uint32_t lds_stride = THREADS * 16u;   // bytes between ring slots

    // prime PIPE-1 buffers
    #pragma unroll
    for (int p = 0; p < PIPE - 1; ++p)
        async_ld_b128(lds_base + (uint32_t)p * lds_stride, src + p * THREADS + tid);

    #pragma unroll 1
    for (int i = 0; i < HITERS; ++i) {
        const int nxt = i + PIPE - 1;
        if (nxt < HITERS) {
            async_ld_b128(lds_base + (uint32_t)(nxt & (PIPE - 1)) * lds_stride,
                          src + nxt * THREADS + tid);
            wait_async_le(PIPE - 1);          // oldest (buffer i) complete
        } else {
            wait_async_le(HITERS - 1 - i);    // drain tail
        }
        f4 v = stage[i & (PIPE - 1)][tid];    // ds_load_b128 from own staged slot
        atomicAdd(&hist[abs_key(v.x) >> 16], 1u);
        atomicAdd(&hist[abs_key(v.y) >> 16], 1u);
        atomicAdd(&hist[abs_key(v.z) >> 16], 1u);
        atomicAdd(&hist[abs_key(v.w) >> 16], 1u);
    }
    __syncthreads();

    uint32_t* __restrict__ gh = g_hist1 + (size_t)row * BINS1;
    for (int b = tid; b < BINS1; b += THREADS) {
        uint32_t c = hist[b];
        if (c) atomicAdd(&gh[b], c);
    }
}

// ---------------- shared select helper (scan histogram from top) ----------------
// Finds highest bin b such that count(bins > b) < k <= count(bins >= b).
// Returns (via thread 0): bin index and k_rem = k - count(bins > b).
__device__ __forceinline__ void select_from_hist(const uint32_t* __restrict__ h,
                                                 int bins, uint32_t k,
                                                 uint32_t* __restrict__ out_bin,
                                                 uint32_t* __restrict__ out_krem) {
    const int BPT = bins / THREADS;           // 128 or 256
    __shared__ uint32_t csum[THREADS];
    __shared__ uint32_t sbins[256];
    __shared__ uint32_t s_tc, s_cum;

    uint32_t s = 0;
    const int base = threadIdx.x * BPT;
    #pragma unroll 8
    for (int j = 0; j < BPT; ++j) s += h[base + j];
    csum[threadIdx.x] = s;
    __syncthreads();

    if (threadIdx.x == 0) {
        uint32_t cum = 0;
        int tc = THREADS - 1;
        for (; tc > 0; --tc) {
            if (cum + csum[tc] >= k) break;
            cum += csum[tc];
        }
        s_tc = (uint32_t)tc;
        s_cum = cum;
    }
    __syncthreads();

    const int tc = (int)s_tc;
    for (int j = threadIdx.x; j < BPT; j += THREADS) sbins[j] = h[tc * BPT + j];
    __syncthreads();

    if (threadIdx.x == 0) {
        uint32_t cum = s_cum;
        int b = BPT - 1;
        for (; b > 0; --b) {
            if (cum + sbins[b] >= k) break;
            cum += sbins[b];
        }
        *out_bin = (uint32_t)(tc * BPT + b);
        *out_krem = k - cum;
    }
}

__global__ void __launch_bounds__(THREADS)
select1_kernel(const uint32_t* __restrict__ g_hist1, const int* __restrict__ kptr,
               uint32_t* __restrict__ meta) {
    const int row = blockIdx.x;
    const uint32_t k = (uint32_t)kptr[0];
    select_from_hist(g_hist1 + (size_t)row * BINS1, BINS1, k,
                     &meta[row * 8 + M_B1], &meta[row * 8 + M_K2]);
}

// ---------------- kernel 3: fine histogram (bits [15:0] within selected bin) ----
__global__ void __launch_bounds__(THREADS)
hist2_kernel(const float* __restrict__ x, const uint32_t* __restrict__ meta,
             uint32_t* __restrict__ g_hist2) {
    __shared__ f4 stage[PIPE][THREADS];

    const int tid = threadIdx.x;
    const int row = blockIdx.y;
    const uint32_t b1 = meta[row * 8 + M_B1];
    uint32_t* __restrict__ gh = g_hist2 + (size_t)row * BINS2;

    const f4* __restrict__ src = (const f4*)x
        + (size_t)row * NV4 + (size_t)blockIdx.x * (THREADS * HITERS);

    const uint32_t lds_base = lds_addr_of(&stage[0][tid]);
    const uint32_t lds_stride = THREADS * 16u;

    #pragma unroll
    for (int p = 0; p < PIPE - 1; ++p)
        async_ld_b128(lds_base + (uint32_t)p * lds_stride, src + p * THREADS + tid);

    #pragma unroll 1
    for (int i = 0; i < HITERS; ++i) {
        const int nxt = i + PIPE - 1;
        if (nxt < HITERS) {
            async_ld_b128(lds_base + (uint32_t)(nxt & (PIPE - 1)) * lds_stride,
                          src + nxt * THREADS + tid);
            wait_async_le(PIPE - 1);
        } else {
            wait_async_le(HITERS - 1 - i);
        }
        f4 v = stage[i & (PIPE - 1)][tid];
        uint32_t k0 = abs_key(v.x), k1 = abs_key(v.y), k2 = abs_key(v.z), k3 = abs_key(v.w);
        if ((k0 >> 16) == b1) atomicAdd(&gh[k0 & 0xffffu], 1u);
        if ((k1 >> 16) == b1) atomicAdd(&gh[k1 & 0xffffu], 1u);
        if ((k2 >> 16) == b1) atomicAdd(&gh[k2 & 0xffffu], 1u);
        if ((k3 >> 16) == b1) atomicAdd(&gh[k3 & 0xffffu], 1u);
    }
}

__global__ void __launch_bounds__(THREADS)
select2_kernel(const uint32_t* __restrict__ g_hist2, uint32_t* __restrict__ meta) {
    const int row = blockIdx.x;
    __shared__ uint32_t s_low, s_krem;
    const uint32_t k2 = meta[row * 8 + M_K2];
    select_from_hist(g_hist2 + (size_t)row * BINS2, BINS2, k2, &s_low, &s_krem);
    __syncthreads();
    if (threadIdx.x == 0) {
        const uint32_t b1 = meta[row * 8 + M_B1];
        meta[row * 8 + M_THR] = (b1 << 16) | s_low;   // exact 31-bit threshold key
        meta[row * 8 + M_KEEPEQ] = s_krem;            // #elements == T to keep
    }
}

// ---------------- kernel 5: thresholded write (streaming, non-temporal) -------
__global__ void __launch_bounds__(THREADS)
write_kernel(const float* __restrict__ x, uint32_t* __restrict__ meta,
             uint32_t* __restrict__ cand, float* __restrict__ out) {
    const int tid = threadIdx.x;
    const int row = blockIdx.y;
    const uint32_t T = meta[row * 8 + M_THR];

    const size_t vbase = (size_t)blockIdx.x * (THREADS * WITERS);
    const f4* __restrict__ src = (const f4*)x + (size_t)row * NV4 + vbase;
    f4* __restrict__ dst = (f4*)out + (size_t)row * NV4 + vbase;
    uint32_t* __restrict__ cc = &meta[row * 8 + M_CANDCNT];
    uint32_t* __restrict__ cb = cand + (size_t)row * CAND_CAP;

    #pragma unroll 1
    for (int i = 0; i < WITERS; i += WUNROLL) {
        f4 v[WUNROLL];
        #pragma unroll
        for (int j = 0; j < WUNROLL; ++j)
            v[j] = __builtin_nontemporal_load(&src[(i + j) * THREADS + tid]);

        #pragma unroll
        for (int j = 0; j < WUNROLL; ++j) {
            const int idx4 = (i + j) * THREADS + tid;
            const uint32_t e0 = (uint32_t)(vbase + idx4) * 4u;
            f4 o;
            #pragma unroll
            for (int c = 0; c < 4; ++c) {
                const float f = v[j][c];
                const uint32_t key = abs_key(f);
                float ov = 0.0f;
                if (key > T) {
                    ov = f;
                } else if (key == T) {
                    uint32_t slot = atomicAdd(cc, 1u);
                    if (slot < CAND_CAP) cb[slot] = e0 + (uint32_t)c;
                }
                o[c] = ov;
            }
            __builtin_nontemporal_store(o, &dst[idx4]);
        }
    }
}

// ---------------- kernel 6: keep lowest-index ties (matches top_k ordering) ----
__global__ void __launch_bounds__(THREADS)
fixup_kernel(const float* __restrict__ x, const uint32_t* __restrict__ meta,
             const uint32_t* __restrict__ cand, float* __restrict__ out) {
    const int row = blockIdx.x;
    __shared__ uint32_t sc[CAND_CAP];         // 16 KB

    uint32_t cnt = meta[row * 8 + M_CANDCNT];
    if (cnt > CAND_CAP) cnt = CAND_CAP;
    const uint32_t keep = meta[row * 8 + M_KEEPEQ];
    const uint32_t* __restrict__ cb = cand + (size_t)row * CAND_CAP;

    for (uint32_t j = threadIdx.x; j < cnt; j += THREADS) sc[j] = cb[j];
    __syncthreads();

    for (uint32_t j = threadIdx.x; j < cnt; j += THREADS) {
        const uint32_t idx = sc[j];
        uint32_t rank = 0;
        for (uint32_t m = 0; m < cnt; ++m) rank += (sc[m] < idx) ? 1u : 0u;
        if (rank < keep) {
            const size_t g = (size_t)row * N_PER_ROW + idx;
            out[g] = x[g];
        }
    }
}

// ---------------- host launcher ----------------
extern "C" void kernel_launch(void* const* d_in, const int* in_sizes, int n_in,
                              void* d_out, int out_size, void* d_ws, size_t ws_size,
                              hipStream_t stream) {
    (void)in_sizes; (void)n_in; (void)out_size; (void)ws_size;

    const float* x   = (const float*)d_in[0];
    const int* kptr  = (const int*)d_in[1];
    float* out       = (float*)d_out;

    uint8_t* ws      = (uint8_t*)d_ws;
    uint32_t* g_hist1 = (uint32_t*)(ws + HIST1_OFF);
    uint32_t* g_hist2 = (uint32_t*)(ws + HIST2_OFF);
    uint32_t* meta    = (uint32_t*)(ws + META_OFF);
    uint32_t* cand    = (uint32_t*)(ws + CAND_OFF);

    hipMemsetAsync(d_ws, 0, WS_ZERO_BYTES, stream);

    hist1_kernel<<<dim3(HBLOCKS_X, BATCH), THREADS, 0, stream>>>(x, g_hist1);
    select1_kernel<<<BATCH, THREADS, 0, stream>>>(g_hist1, kptr, meta);
    hist2_kernel<<<dim3(HBLOCKS_X, BATCH), THREADS, 0, stream>>>(x, meta, g_hist2);
    select2_kernel<<<BATCH, THREADS, 0, stream>>>(g_hist2, meta);
    write_kernel<<<dim3(WBLOCKS_X, BATCH), THREADS, 0, stream>>>(x, meta, cand, out);
    fixup_kernel<<<BATCH, THREADS, 0, stream>>>(x, meta, cand, out);
}